// RenderLayer_69733089017977
// MI455X (gfx1250) — compile-verified
//
#include <hip/hip_runtime.h>
#include <cstdint>

// ---------------- problem constants (match reference) ----------------
constexpr int   kH  = 384,  kW  = 384;
constexpr int   kTH = 1024, kTW = 1024;
constexpr int   kN  = 2,    kV  = 2048, kF = 2048;
constexpr float kEPS = 1e-8f;

constexpr int kChunk   = 256;          // triangles staged in LDS at a time
constexpr int kLdsStr  = 18;           // floats per tri in LDS (16 data + 2 pad, keeps b64 aligned, 2-way banks)
constexpr int kStripsX = kW / 16;      // 24 strips of 16 px per row
constexpr int kSPI     = kStripsX * kH;// strips per image = 9216

typedef __attribute__((ext_vector_type(2))) float v2f;
typedef __attribute__((ext_vector_type(8))) float v8f;

// ---------------- kernel 1: vertex projection ----------------
__global__ void project_kernel(const float* __restrict__ v,
                               const float* __restrict__ campos,
                               const float* __restrict__ camrot,
                               const float* __restrict__ focal,
                               const float* __restrict__ princpt,
                               float* __restrict__ vpix) {
  int t = blockIdx.x * blockDim.x + threadIdx.x;
  if (t >= kN * kV) return;
  int n = t / kV;
  const float* vv = v + (size_t)t * 3;
  const float* cp = campos + n * 3;
  const float* R  = camrot + n * 9;
  float dx = vv[0] - cp[0], dy = vv[1] - cp[1], dz = vv[2] - cp[2];
  float X = R[0]*dx + R[1]*dy + R[2]*dz;
  float Y = R[3]*dx + R[4]*dy + R[5]*dz;
  float Z = R[6]*dx + R[7]*dy + R[8]*dz;
  const float* Fm = focal + n * 4;
  const float* pp = princpt + n * 2;
  float xz = X / Z, yz = Y / Z;
  vpix[(size_t)t*3 + 0] = Fm[0]*xz + Fm[1]*yz + pp[0];
  vpix[(size_t)t*3 + 1] = Fm[2]*xz + Fm[3]*yz + pp[1];
  vpix[(size_t)t*3 + 2] = Z;
}

// ---------------- kernel 2: per-triangle affine edge/zinv coefficients ----------------
// For triangle f: b_i(px,py) = P_i*px + Q_i*py + R_i  (inv=1/area folded in),
// zinv(px,py)    = Pz*px + Qz*py + Rz.
// Layout per tri (16 floats): P0,Q0,R0,0, P1,Q1,R1,0, P2,Q2,R2,0, Pz,Qz,Rz,0
__global__ void coef_kernel(const float* __restrict__ vpix,
                            const int* __restrict__ vi,
                            float* __restrict__ coef) {
  int t = blockIdx.x * blockDim.x + threadIdx.x;
  if (t >= kN * kF) return;
  int n = t / kF, f = t % kF;
  int i0 = vi[f*3+0], i1 = vi[f*3+1], i2 = vi[f*3+2];
  const float* p0 = vpix + ((size_t)n*kV + i0) * 3;
  const float* p1 = vpix + ((size_t)n*kV + i1) * 3;
  const float* p2 = vpix + ((size_t)n*kV + i2) * 3;
  float x0=p0[0], y0=p0[1], z0=p0[2];
  float x1=p1[0], y1=p1[1], z1=p1[2];
  float x2=p2[0], y2=p2[1], z2=p2[2];

  float area = (x1-x0)*(y2-y0) - (y1-y0)*(x2-x0);
  bool  ok   = fabsf(area) > 1e-9f;
  float inv  = 1.0f / (ok ? area : 1.0f);

  float P0 = -(y2-y1)*inv, Q0 = (x2-x1)*inv, R0 = ((y2-y1)*x1 - (x2-x1)*y1)*inv;
  float P1 = -(y0-y2)*inv, Q1 = (x0-x2)*inv, R1 = ((y0-y2)*x2 - (x0-x2)*y2)*inv;
  float P2 = -(y1-y0)*inv, Q2 = (x1-x0)*inv, R2 = ((y1-y0)*x0 - (x1-x0)*y0)*inv;
  float rz0 = 1.0f/z0, rz1 = 1.0f/z1, rz2 = 1.0f/z2;
  float Pz = P0*rz0 + P1*rz1 + P2*rz2;
  float Qz = Q0*rz0 + Q1*rz1 + Q2*rz2;
  float Rz = R0*rz0 + R1*rz1 + R2*rz2;
  if (!ok) { // degenerate: force "outside" everywhere
    P0 = Q0 = 0.f; R0 = -1.f;
    P1 = Q1 = R1 = 0.f; P2 = Q2 = R2 = 0.f; Pz = Qz = Rz = 0.f;
  }
  float4* o = (float4*)(coef + (size_t)t * 16);
  o[0] = make_float4(P0, Q0, R0, 0.f);
  o[1] = make_float4(P1, Q1, R1, 0.f);
  o[2] = make_float4(P2, Q2, R2, 0.f);
  o[3] = make_float4(Pz, Qz, Rz, 0.f);
}

// ---------------- kernel 3: WMMA rasterizer (depth/index buffer) ----------------
// One wave32 owns a 16x1 pixel strip. Per 16-triangle group, 4x
// V_WMMA_F32_16X16X4_F32 evaluate b0,b1,b2,zinv for 16 tri x 16 px.
// Winner per pixel: max zinv (== min depth), ties -> smallest face id
// (matches the reference scan + argmin semantics). Inside test folded to
// v_min3_f32 + one compare.
__global__ __launch_bounds__(128) void raster_kernel(const float* __restrict__ coef,
                                                     int* __restrict__ idxbuf) {
  const int lane = threadIdx.x & 31;
  const int wave = threadIdx.x >> 5;
  const int sid  = blockIdx.x * 4 + wave;       // strip id
  const int n    = sid / kSPI;
  const int rem  = sid % kSPI;
  const int row  = rem / kStripsX;
  const int tx   = rem % kStripsX;
  const int lm   = lane & 15;

  const float pxf = (float)(tx * 16 + lm) + 0.5f;
  const float pyf = (float)row + 0.5f;
  // B operand (4x16): K0=px, K1=py (lanes 0-15); K2=1, K3=0 (lanes 16-31)
  const v2f bmat = (lane < 16) ? (v2f){pxf, pyf} : (v2f){1.0f, 0.0f};

  __shared__ __align__(16) float sc[kChunk * kLdsStr];

  unsigned long long best = 0ull;
  const float* gbase = coef + (size_t)n * kF * 16;

  for (int c = 0; c < kF / kChunk; ++c) {
    __syncthreads();
    const float* cb = gbase + (size_t)c * kChunk * 16;
    for (int i = threadIdx.x; i < kChunk * 16; i += 128)
      sc[(i >> 4) * kLdsStr + (i & 15)] = cb[i];
    __syncthreads();
    if (c + 1 < kF / kChunk)  // global_prefetch_b8 of next coefficient tile
      __builtin_prefetch(cb + kChunk * 16 + threadIdx.x * 32, 0, 0);

    for (int g = 0; g < kChunk / 16; ++g) {
      // A operand: lanes 0-15 read (P,Q)=K0,K1; lanes 16-31 read (R,0)=K2,K3
      const float* tp = sc + (g * 16 + lm) * kLdsStr + ((lane >> 4) << 1);
      v2f a0 = *(const v2f*)(tp + 0);
      v2f a1 = *(const v2f*)(tp + 4);
      v2f a2 = *(const v2f*)(tp + 8);
      v2f az = *(const v2f*)(tp + 12);
      v8f zacc = {0.f,0.f,0.f,0.f,0.f,0.f,0.f,0.f};
      v8f d0 = __builtin_amdgcn_wmma_f32_16x16x4_f32(false, a0, false, bmat, (short)0, zacc, false, false);
      v8f d1 = __builtin_amdgcn_wmma_f32_16x16x4_f32(false, a1, false, bmat, (short)0, zacc, false, false);
      v8f d2 = __builtin_amdgcn_wmma_f32_16x16x4_f32(false, a2, false, bmat, (short)0, zacc, false, false);
      v8f dz = __builtin_amdgcn_wmma_f32_16x16x4_f32(false, az, false, bmat, (short)0, zacc, false, false);

      // D layout: lanes 0-15 -> M=r, lanes 16-31 -> M=r+8 (same pixel column)
      const int mb = c * kChunk + g * 16 + ((lane >> 4) << 3);
#pragma unroll
      for (int r = 0; r < 8; ++r) {
        float zv = dz[r];
        // inside test as a single v_min3_f32 + compare; zv>EPS also rejects
        // NaN/degenerate rows (matches reference comparison semantics)
        float mn = fminf(fminf(d0[r], d1[r]), d2[r]);
        bool cand = (mn >= 0.f) && (zv > kEPS);
        unsigned zb = __float_as_uint(zv); // zv>0 -> monotone as uint
        unsigned long long key =
            ((unsigned long long)zb << 32) | (unsigned)(0x7FFFFFFF - (mb + r));
        best = (cand && key > best) ? key : best;
      }
    }
  }

  // merge triangle halves: lane L holds pixel L%16 for 8 of 16 rows
  unsigned lo = (unsigned)best, hi = (unsigned)(best >> 32);
  unsigned olo = (unsigned)__shfl_xor((int)lo, 16, 32);
  unsigned ohi = (unsigned)__shfl_xor((int)hi, 16, 32);
  unsigned long long other = ((unsigned long long)ohi << 32) | olo;
  if (other > best) best = other;

  if (lane < 16) {
    int fid = best ? (int)(0x7FFFFFFFu - (unsigned)best) : -1;
    idxbuf[((size_t)n * kH + row) * kW + tx * 16 + lm] = fid;
  }
}

// ---------------- kernel 4: bary recompute + UV interp + bilinear sample ----------------
__global__ void render_kernel(const float* __restrict__ vpix,
                              const int* __restrict__ vi,
                              const int* __restrict__ vti,
                              const float* __restrict__ vt,
                              const float* __restrict__ tex,
                              const int* __restrict__ idxbuf,
                              float* __restrict__ out) {
  int t = blockIdx.x * blockDim.x + threadIdx.x;
  if (t >= kN * kH * kW) return;
  int n = t / (kH * kW);
  int p = t % (kH * kW);
  int h = p / kW, w = p % kW;

  int idx = idxbuf[t];
  float m = (idx >= 0) ? 1.0f : 0.0f;
  int f = max(idx, 0);

  int i0 = vi[f*3+0], i1 = vi[f*3+1], i2 = vi[f*3+2];
  const float* p0 = vpix + ((size_t)n*kV + i0) * 3;
  const float* p1 = vpix + ((size_t)n*kV + i1) * 3;
  const float* p2 = vpix + ((size_t)n*kV + i2) * 3;
  float x0=p0[0], y0=p0[1], z0=p0[2];
  float x1=p1[0], y1=p1[1], z1=p1[2];
  float x2=p2[0], y2=p2[1], z2=p2[2];
  float px = (float)w + 0.5f, py = (float)h + 0.5f;

  // exact reference _bary formulas
  float area = (x1-x0)*(y2-y0) - (y1-y0)*(x2-x0);
  bool  ok   = fabsf(area) > 1e-9f;
  float inv  = 1.0f / (ok ? area : 1.0f);
  float b0 = ((x2-x1)*(py-y1) - (y2-y1)*(px-x1)) * inv;
  float b1 = ((x0-x2)*(py-y2) - (y0-y2)*(px-x2)) * inv;
  float b2 = ((x1-x0)*(py-y0) - (y1-y0)*(px-x0)) * inv;

  float w0 = b0/z0, w1 = b1/z1, w2 = b2/z2;
  float zinv = fmaxf(w0 + w1 + w2, kEPS);
  float c0 = m * (w0 / zinv), c1 = m * (w1 / zinv), c2 = m * (w2 / zinv);

  int j0 = vti[f*3+0], j1 = vti[f*3+1], j2 = vti[f*3+2];
  float u0 = 2.f*vt[j0*2+0]-1.f, v0 = 2.f*vt[j0*2+1]-1.f;
  float u1 = 2.f*vt[j1*2+0]-1.f, v1 = 2.f*vt[j1*2+1]-1.f;
  float u2 = 2.f*vt[j2*2+0]-1.f, v2 = 2.f*vt[j2*2+1]-1.f;
  float gu = c0*u0 + c1*u1 + c2*u2;
  float gv = c0*v0 + c1*v1 + c2*v2;

  float ix = ((gu + 1.0f) * kTW - 1.0f) * 0.5f;
  float iy = ((gv + 1.0f) * kTH - 1.0f) * 0.5f;
  float ix0 = floorf(ix), iy0 = floorf(iy);
  float wx = ix - ix0, wy = iy - iy0;
  int xa = min(max((int)ix0,     0), kTW-1);
  int xb = min(max((int)ix0 + 1, 0), kTW-1);
  int ya = min(max((int)iy0,     0), kTH-1);
  int yb = min(max((int)iy0 + 1, 0), kTH-1);

  float wa = (1.f-wx)*(1.f-wy), wb = wx*(1.f-wy), wc = (1.f-wx)*wy, wd = wx*wy;
#pragma unroll
  for (int c = 0; c < 3; ++c) {
    const float* tb = tex + ((size_t)n*3 + c) * (size_t)kTH * kTW;
    float val = tb[(size_t)ya*kTW + xa]*wa + tb[(size_t)ya*kTW + xb]*wb
              + tb[(size_t)yb*kTW + xa]*wc + tb[(size_t)yb*kTW + xb]*wd;
    out[(((size_t)n*3 + c) * kH + h) * kW + w] = val * m;
  }
}

// ---------------- launcher ----------------
extern "C" void kernel_launch(void* const* d_in, const int* in_sizes, int n_in,
                              void* d_out, int out_size, void* d_ws, size_t ws_size,
                              hipStream_t stream) {
  (void)in_sizes; (void)n_in; (void)out_size; (void)ws_size;
  const float* v       = (const float*)d_in[0];
  const float* tex     = (const float*)d_in[1];
  const float* campos  = (const float*)d_in[2];
  const float* camrot  = (const float*)d_in[3];
  const float* focal   = (const float*)d_in[4];
  const float* princpt = (const float*)d_in[5];
  const float* vt      = (const float*)d_in[6];
  const int*   vi      = (const int*)d_in[7];
  const int*   vti     = (const int*)d_in[8];
  float* out = (float*)d_out;

  // workspace layout
  float* vpix = (float*)d_ws;                       // N*V*3
  float* coef = vpix + (size_t)kN * kV * 3;         // N*F*16
  int*   idxb = (int*)(coef + (size_t)kN * kF * 16);// N*H*W

  project_kernel<<<(kN*kV + 255)/256, 256, 0, stream>>>(v, campos, camrot, focal, princpt, vpix);
  coef_kernel<<<(kN*kF + 255)/256, 256, 0, stream>>>(vpix, vi, coef);
  raster_kernel<<<(kN * kSPI) / 4, 128, 0, stream>>>(coef, idxb);
  render_kernel<<<(kN*kH*kW + 255)/256, 256, 0, stream>>>(vpix, vi, vti, vt, tex, idxb, out);
}